// ViewRenderingCrossView_37855841747696
// MI455X (gfx1250) — compile-verified
//
#include <hip/hip_runtime.h>
#include <hip/hip_bf16.h>

#define BB 8
#define CC 3
#define HH 512
#define WW 512
#define HWN (HH * WW)

typedef float v2f __attribute__((ext_vector_type(2)));
typedef float v8f __attribute__((ext_vector_type(8)));

// ---------------------------------------------------------------------------
// Setup kernel (1 wave): compute per-batch fused transform
//   cam_s = A * [u*z, v*z, z] + c
// where M = R_s * R_t^T  (done via V_WMMA_F32_16X16X4_F32, block-diag packing),
//       A = M * Dinv(K_tgt),  c = t_s - M * t_t.
// Params per batch (16 floats): A[9], c[3], fx_s, fy_s, cx_s, cy_s.
// ---------------------------------------------------------------------------
__global__ __launch_bounds__(32) void setup_params_kernel(
    const float* __restrict__ Ksrc, const float* __restrict__ Ktgt,
    const float* __restrict__ Esrc, const float* __restrict__ Etgt,
    float* __restrict__ params)
{
    __shared__ float Mc[BB * 9];
    const int lane = threadIdx.x;       // 0..31 (wave32)
    const int half = lane >> 4;         // 0 or 1
    const int lm   = lane & 15;

    // Two WMMA passes: batches 0..4 packed as 5 3-row blocks, then batches 5..7.
    for (int g = 0; g < 2; ++g) {
        const int base = g * 5;
        const int nb   = (g == 0) ? 5 : 3;

        v2f a;  a[0] = 0.0f;  a[1] = 0.0f;     // A: 16x4, rows = stacked R_s
        v2f bm; bm[0] = 0.0f; bm[1] = 0.0f;    // B: 4x16, cols = stacked R_t^T
        #pragma unroll
        for (int j = 0; j < 2; ++j) {
            const int K = 2 * half + j;        // K index per ISA 16x4 layout
            if (K < 3 && lm < 3 * nb) {
                const int bb = base + lm / 3;
                const int r  = lm % 3;
                a[j]  = Esrc[bb * 16 + r * 4 + K];   // A[m][k] = R_s[r][k]
                bm[j] = Etgt[bb * 16 + r * 4 + K];   // B[k][n] = R_t^T[k][c] = R_t[c][k]
            }
        }

        v8f acc = {};
        acc = __builtin_amdgcn_wmma_f32_16x16x4_f32(
            /*neg_a=*/false, a, /*neg_b=*/false, bm,
            /*c_mod=*/(short)0, acc, /*reuse_a=*/false, /*reuse_b=*/false);

        // Extract diagonal 3x3 blocks: M_b = R_s,b * R_t,b^T
        #pragma unroll
        for (int v = 0; v < 8; ++v) {
            const int Mrow = v + 8 * half;     // C/D layout: M = vgpr + 8*(lane/16)
            const int N    = lm;               // N = lane % 16
            if (Mrow < 3 * nb) {
                const int blk = Mrow / 3;
                if (N >= 3 * blk && N < 3 * blk + 3) {
                    Mc[(base + blk) * 9 + (Mrow % 3) * 3 + (N - 3 * blk)] = acc[v];
                }
            }
        }
    }
    __syncthreads();

    if (lane < BB) {
        const int b = lane;
        const float* M = &Mc[b * 9];

        const float fx_t = Ktgt[b * 9 + 0], fy_t = Ktgt[b * 9 + 4];
        const float cx_t = Ktgt[b * 9 + 2], cy_t = Ktgt[b * 9 + 5];
        const float ifx = 1.0f / (fx_t + 1e-8f);
        const float ify = 1.0f / (fy_t + 1e-8f);

        // A = M * Dinv,  Dinv = [[ifx,0,-cx*ifx],[0,ify,-cy*ify],[0,0,1]]
        float A[9];
        #pragma unroll
        for (int r = 0; r < 3; ++r) {
            A[r * 3 + 0] = M[r * 3 + 0] * ifx;
            A[r * 3 + 1] = M[r * 3 + 1] * ify;
            A[r * 3 + 2] = -cx_t * ifx * M[r * 3 + 0]
                           - cy_t * ify * M[r * 3 + 1]
                           + M[r * 3 + 2];
        }
        // c = t_s - M * t_t
        const float ttx = Etgt[b * 16 + 3], tty = Etgt[b * 16 + 7], ttz = Etgt[b * 16 + 11];
        const float c0 = Esrc[b * 16 + 3]  - (M[0] * ttx + M[1] * tty + M[2] * ttz);
        const float c1 = Esrc[b * 16 + 7]  - (M[3] * ttx + M[4] * tty + M[5] * ttz);
        const float c2 = Esrc[b * 16 + 11] - (M[6] * ttx + M[7] * tty + M[8] * ttz);

        float* p = params + b * 16;
        #pragma unroll
        for (int i = 0; i < 9; ++i) p[i] = A[i];
        p[9] = c0; p[10] = c1; p[11] = c2;
        p[12] = Ksrc[b * 9 + 0];   // fx_s
        p[13] = Ksrc[b * 9 + 4];   // fy_s
        p[14] = Ksrc[b * 9 + 2];   // cx_s
        p[15] = Ksrc[b * 9 + 5];   // cy_s
    }
}

// ---------------------------------------------------------------------------
// Main warp kernel: 1 pixel per thread, batch index uniform per block so the
// 16 params land in SGPRs. Memory-bound: coalesced depth read + 12 cached
// gathers + 4 coalesced stores per thread.
// ---------------------------------------------------------------------------
__global__ __launch_bounds__(256) void warp_kernel(
    const float* __restrict__ src, const float* __restrict__ depth,
    const float* __restrict__ params, float* __restrict__ out)
{
    const int blocksPerBatch = HWN / 256;                 // 1024
    const int b   = blockIdx.x / blocksPerBatch;          // uniform (SGPR)
    const int pix = (blockIdx.x % blocksPerBatch) * 256 + threadIdx.x;

    const float* __restrict__ p = params + b * 16;
    const float A00 = p[0], A01 = p[1], A02 = p[2];
    const float A10 = p[3], A11 = p[4], A12 = p[5];
    const float A20 = p[6], A21 = p[7], A22 = p[8];
    const float c0  = p[9], c1  = p[10], c2 = p[11];
    const float fxs = p[12], fys = p[13], cxs = p[14], cys = p[15];

    const int yy = pix / WW;
    const int xx = pix % WW;

    const float z  = depth[b * HWN + pix];
    const float uz = (float)xx * z;
    const float vz = (float)yy * z;

    const float X  = fmaf(A00, uz, fmaf(A01, vz, fmaf(A02, z, c0)));
    const float Y  = fmaf(A10, uz, fmaf(A11, vz, fmaf(A12, z, c1)));
    const float Zr = fmaf(A20, uz, fmaf(A21, vz, fmaf(A22, z, c2)));
    const float Z  = fmaxf(Zr, 1e-4f);
    const float rz = 1.0f / Z;

    const float us = fmaf(fxs, X * rz, cxs);
    const float vs = fmaf(fys, Y * rz, cys);

    const float un = 2.0f * (us / (WW - 1.0f)) - 1.0f;
    const float vn = 2.0f * (vs / (HH - 1.0f)) - 1.0f;

    // bilinear with zeros padding (validity folded into weights)
    const float x0f = floorf(us), y0f = floorf(vs);
    const float x1f = x0f + 1.0f, y1f = y0f + 1.0f;
    const float wx1 = us - x0f, wx0 = 1.0f - wx1;
    const float wy1 = vs - y0f, wy0 = 1.0f - wy1;

    const bool vx0 = (x0f >= 0.0f) && (x0f <= (float)(WW - 1));
    const bool vx1 = (x1f >= 0.0f) && (x1f <= (float)(WW - 1));
    const bool vy0 = (y0f >= 0.0f) && (y0f <= (float)(HH - 1));
    const bool vy1 = (y1f >= 0.0f) && (y1f <= (float)(HH - 1));

    const int ix0 = (int)fminf(fmaxf(x0f, 0.0f), (float)(WW - 1));
    const int ix1 = (int)fminf(fmaxf(x1f, 0.0f), (float)(WW - 1));
    const int iy0 = (int)fminf(fmaxf(y0f, 0.0f), (float)(HH - 1));
    const int iy1 = (int)fminf(fmaxf(y1f, 0.0f), (float)(HH - 1));

    const float w00 = (vx0 && vy0) ? wx0 * wy0 : 0.0f;
    const float w10 = (vx1 && vy0) ? wx1 * wy0 : 0.0f;
    const float w01 = (vx0 && vy1) ? wx0 * wy1 : 0.0f;
    const float w11 = (vx1 && vy1) ? wx1 * wy1 : 0.0f;

    const int i00 = iy0 * WW + ix0;
    const int i10 = iy0 * WW + ix1;
    const int i01 = iy1 * WW + ix0;
    const int i11 = iy1 * WW + ix1;

    const float* __restrict__ sb = src + (size_t)b * CC * HWN;
    #pragma unroll
    for (int ch = 0; ch < CC; ++ch) {
        const float* __restrict__ sp = sb + ch * HWN;
        const float val = sp[i00] * w00 + sp[i10] * w10
                        + sp[i01] * w01 + sp[i11] * w11;
        out[((size_t)b * CC + ch) * HWN + pix] = val;
    }

    const bool okm = (un >= -1.0f) && (un <= 1.0f) &&
                     (vn >= -1.0f) && (vn <= 1.0f) && (Zr > 1e-4f);
    out[(size_t)BB * CC * HWN + (size_t)b * HWN + pix] = okm ? 1.0f : 0.0f;
}

extern "C" void kernel_launch(void* const* d_in, const int* in_sizes, int n_in,
                              void* d_out, int out_size, void* d_ws, size_t ws_size,
                              hipStream_t stream) {
    const float* src   = (const float*)d_in[0];  // (8,3,512,512)
    const float* depth = (const float*)d_in[1];  // (8,1,512,512)
    const float* Ksrc  = (const float*)d_in[2];  // (8,3,3)
    const float* Ktgt  = (const float*)d_in[3];  // (8,3,3)
    const float* Esrc  = (const float*)d_in[4];  // (8,4,4)
    const float* Etgt  = (const float*)d_in[5];  // (8,4,4)
    float* out    = (float*)d_out;               // warped (8*3*HW) ++ mask (8*HW)
    float* params = (float*)d_ws;                // 8*16 floats scratch

    setup_params_kernel<<<dim3(1), dim3(32), 0, stream>>>(Ksrc, Ktgt, Esrc, Etgt, params);
    warp_kernel<<<dim3(BB * (HWN / 256)), dim3(256), 0, stream>>>(src, depth, params, out);
}